// MeshConvTransposeGNReLU_20761871909670
// MI455X (gfx1250) — compile-verified
//
#include <hip/hip_runtime.h>
#include <hip/hip_bf16.h>

// ---------------- problem constants (from reference) ----------------
#define NVV    40962     // vertices (output)
#define NVP    10242     // vertices (input, rest zero-padded)
#define CIN    64
#define COUT   64
#define BATCH  8
#define KNBR   7
#define NGROUP 32        // GroupNorm groups -> 2 channels per group
#define EPSGN  1e-05f

#define VPB    128                 // vertices per block (8 waves x 16)
#define NTILE  ((NVV + VPB - 1) / VPB)   // 321 vertex tiles

typedef float v2f __attribute__((ext_vector_type(2)));
typedef float v8f __attribute__((ext_vector_type(8)));

// ---------------------------------------------------------------
// K1: transpose x (B,Cin,NVP) -> xT (B,NVP,Cin) so that per-vertex
// channel gathers are 256B-contiguous.
// ---------------------------------------------------------------
__global__ __launch_bounds__(256) void k_transpose(const float* __restrict__ x,
                                                   float* __restrict__ xT) {
  __shared__ float tile[16][65];
  const int vbase = blockIdx.x * 16;
  const int b     = blockIdx.y;
  const int tid   = threadIdx.x;
#pragma unroll
  for (int p = 0; p < 4; ++p) {
    int idx = p * 256 + tid;
    int v = idx & 15, i = idx >> 4;
    if (vbase + v < NVP)
      tile[v][i] = x[((size_t)b * CIN + i) * NVP + vbase + v];
  }
  __syncthreads();
#pragma unroll
  for (int p = 0; p < 4; ++p) {
    int idx = p * 256 + tid;
    int i = idx & 63, v = idx >> 6;
    if (vbase + v < NVP)
      xT[((size_t)b * NVP + vbase + v) * CIN + i] = tile[v][i];
  }
}

// ---------------------------------------------------------------
// K2: fused gather + feature build + fp32 WMMA GEMM (+bias) + GN partials.
//   out_pre[b,o,v] = sum_{i,k} coeffs[o,i,k] * feat[b,i,k,v] + bias[o]
// A matrix = coeffs flat [o][i*4+k]  (64 x 256)
// B matrix = feat tile               (256 x 16 vertices)
// 64 K-steps of v_wmma_f32_16x16x4_f32 x 4 M-tiles per wave.
// ---------------------------------------------------------------
__global__ __launch_bounds__(256) void k_main(const float* __restrict__ xT,
                                              const float* __restrict__ Lv,
                                              const float* __restrict__ EWv,
                                              const float* __restrict__ NSv,
                                              const float* __restrict__ coeffs,
                                              const float* __restrict__ bias,
                                              const int*   __restrict__ nbr,
                                              float* __restrict__ out,
                                              float* __restrict__ partials) {
  extern __shared__ float smem[];
  float* sA     = smem;                  // 64*256 = 16384 floats (64 KB)
  float* sF     = smem + 16384;          // 8 waves * 4096 floats (128 KB)
  float* sStats = smem + 16384 + 32768;  // 32 groups * {sum, sumsq}

  const int tid  = threadIdx.x;
  const int wave = tid >> 5;
  const int lane = tid & 31;
  const int b    = blockIdx.y;
  const int vblk = blockIdx.x * VPB;
  const int vbase = vblk + wave * 16;

  // stage A (coeffs) into LDS: flat copy, coeffs[o][i][k] == A[o][i*4+k]
  for (int p = tid; p < COUT * 256; p += 256) sA[p] = coeffs[p];
  if (tid < 2 * NGROUP) sStats[tid] = 0.0f;
  __syncthreads();

  // ---- build feat tile: sFw[v][col], col = i*4 + {up,lap,ew,ns} ----
  float* sFw = sF + wave * 4096;
  const float* xb = xT + (size_t)b * NVP * CIN;
  for (int v = 0; v < 16; ++v) {
    const int vg = vbase + v;               // wave-uniform
    float up0 = 0.f, lap0 = 0.f, ew0 = 0.f, ns0 = 0.f;
    float up1 = 0.f, lap1 = 0.f, ew1 = 0.f, ns1 = 0.f;
    if (vg < NVV) {
      if (vg < NVP) {
        up0 = xb[(size_t)vg * CIN + lane];
        up1 = xb[(size_t)vg * CIN + lane + 32];
      }
#pragma unroll
      for (int j = 0; j < KNBR; ++j) {
        const int   n  = nbr[vg * KNBR + j];   // uniform across wave
        const float wl = Lv [vg * KNBR + j];
        const float we = EWv[vg * KNBR + j];
        const float wn = NSv[vg * KNBR + j];
        if (n < NVP) {                         // padded region contributes 0
          const float g0 = xb[(size_t)n * CIN + lane];
          const float g1 = xb[(size_t)n * CIN + lane + 32];
          lap0 = fmaf(g0, wl, lap0); ew0 = fmaf(g0, we, ew0); ns0 = fmaf(g0, wn, ns0);
          lap1 = fmaf(g1, wl, lap1); ew1 = fmaf(g1, we, ew1); ns1 = fmaf(g1, wn, ns1);
        }
      }
    }
    float4* row = (float4*)(sFw + v * 256);
    row[lane]      = make_float4(up0, lap0, ew0, ns0);  // cols 4*lane..+3
    row[lane + 32] = make_float4(up1, lap1, ew1, ns1);  // cols 4*(lane+32)..
  }
  __syncthreads();

  // ---- GEMM: D = A(64x256) * B(256x16), K-chunks of 4 ----
  // A frag (16x4 f32): lanes0-15 -> v0=K0,v1=K1 ; lanes16-31 -> v0=K2,v1=K3
  // B frag (4x16 f32): mirrored; C/D: lane<16 M=r, lane>=16 M=r+8, N=lane&15
  v8f acc[4] = {};
  const int half = lane >> 4;
  const int l16  = lane & 15;
#pragma unroll 4
  for (int ks = 0; ks < 64; ++ks) {
    const int kcol = 4 * ks + 2 * half;
    const v2f bf = *(const v2f*)(sFw + l16 * 256 + kcol);     // 8B-aligned pair
    const float* ap = sA + l16 * 256 + kcol;
    const v2f a0 = *(const v2f*)(ap);
    const v2f a1 = *(const v2f*)(ap + 4096);
    const v2f a2 = *(const v2f*)(ap + 8192);
    const v2f a3 = *(const v2f*)(ap + 12288);
    acc[0] = __builtin_amdgcn_wmma_f32_16x16x4_f32(false, a0, false, bf, (short)0, acc[0], false, false);
    acc[1] = __builtin_amdgcn_wmma_f32_16x16x4_f32(false, a1, false, bf, (short)0, acc[1], false, false);
    acc[2] = __builtin_amdgcn_wmma_f32_16x16x4_f32(false, a2, false, bf, (short)0, acc[2], false, false);
    acc[3] = __builtin_amdgcn_wmma_f32_16x16x4_f32(false, a3, false, bf, (short)0, acc[3], false, false);
  }

  // ---- epilogue: bias, store pre-GN, accumulate GN partials ----
  const int  vg     = vblk + wave * 16 + l16;
  const bool vvalid = vg < NVV;
#pragma unroll
  for (int m = 0; m < 4; ++m) {
#pragma unroll
    for (int r = 0; r < 8; ++r) {
      const int o = m * 16 + 8 * half + r;
      const float val = acc[m][r] + bias[o];
      if (vvalid) {
        out[((size_t)b * COUT + o) * NVV + vg] = val;
        atomicAdd(&sStats[(o >> 1) * 2],     val);        // ds_add_f32
        atomicAdd(&sStats[(o >> 1) * 2 + 1], val * val);
      }
    }
  }
  __syncthreads();
  if (tid < 2 * NGROUP)
    partials[((size_t)(b * NTILE + blockIdx.x)) * (2 * NGROUP) + tid] = sStats[tid];
}

// ---------------------------------------------------------------
// K3: reduce per-tile partials -> mean/rstd per (b, group)
// ---------------------------------------------------------------
__global__ void k_stats(const float* __restrict__ partials, float* __restrict__ mstd) {
  const int g = blockIdx.x, b = blockIdx.y, lane = threadIdx.x;  // 32 threads
  float s = 0.f, sq = 0.f;
  for (int t = lane; t < NTILE; t += 32) {
    const float* p = partials + ((size_t)(b * NTILE + t)) * (2 * NGROUP) + g * 2;
    s += p[0]; sq += p[1];
  }
#pragma unroll
  for (int off = 16; off > 0; off >>= 1) {
    s  += __shfl_xor(s, off);
    sq += __shfl_xor(sq, off);
  }
  if (lane == 0) {
    const float cnt  = (float)(2 * NVV);   // (Cout/GROUPS)*NV elements per group
    const float mean = s / cnt;
    const float var  = sq / cnt - mean * mean;
    mstd[(b * NGROUP + g) * 2 + 0] = mean;
    mstd[(b * NGROUP + g) * 2 + 1] = rsqrtf(var + EPSGN);
  }
}

// ---------------------------------------------------------------
// K4: normalize + affine + ReLU (in-place on d_out)
// ---------------------------------------------------------------
__global__ __launch_bounds__(256) void k_norm(float* __restrict__ out,
                                              const float* __restrict__ mstd,
                                              const float* __restrict__ gamma,
                                              const float* __restrict__ beta) {
  const size_t total = (size_t)BATCH * COUT * NVV;
  size_t i = (size_t)blockIdx.x * blockDim.x + threadIdx.x;
  if (i >= total) return;
  const int o = (int)((i / NVV) % COUT);
  const int b = (int)(i / ((size_t)NVV * COUT));
  const int g = o >> 1;
  const float mean = mstd[(b * NGROUP + g) * 2 + 0];
  const float rstd = mstd[(b * NGROUP + g) * 2 + 1];
  float v = out[i];
  v = (v - mean) * rstd * gamma[o] + beta[o];
  out[i] = fmaxf(v, 0.0f);
}

// ---------------------------------------------------------------
extern "C" void kernel_launch(void* const* d_in, const int* in_sizes, int n_in,
                              void* d_out, int out_size, void* d_ws, size_t ws_size,
                              hipStream_t stream) {
  const float* x      = (const float*)d_in[0];
  const float* Lv     = (const float*)d_in[1];
  const float* EWv    = (const float*)d_in[2];
  const float* NSv    = (const float*)d_in[3];
  const float* coeffs = (const float*)d_in[4];
  const float* bias   = (const float*)d_in[5];
  const float* gamma  = (const float*)d_in[6];
  const float* beta   = (const float*)d_in[7];
  const int*   nbr    = (const int*)d_in[8];
  float* out = (float*)d_out;

  // workspace layout: xT | partials | mstd   (~21.6 MB total)
  float* ws       = (float*)d_ws;
  float* xT       = ws;                                        // B*NVP*CIN
  float* partials = xT + (size_t)BATCH * NVP * CIN;            // B*NTILE*64
  float* mstd     = partials + (size_t)BATCH * NTILE * 2 * NGROUP;  // B*32*2

  dim3 gT((NVP + 15) / 16, BATCH);
  k_transpose<<<gT, 256, 0, stream>>>(x, xT);

  dim3 gM(NTILE, BATCH);
  const size_t lds = (size_t)(16384 + 8 * 4096 + 2 * NGROUP) * sizeof(float);
  k_main<<<gM, 256, lds, stream>>>(xT, Lv, EWv, NSv, coeffs, bias, nbr, out, partials);

  dim3 gS(NGROUP, BATCH);
  k_stats<<<gS, 32, 0, stream>>>(partials, mstd);

  const size_t total = (size_t)BATCH * COUT * NVV;
  k_norm<<<(unsigned)((total + 255) / 256), 256, 0, stream>>>(out, mstd, gamma, beta);
}